// SMPL_88811333747290
// MI455X (gfx1250) — compile-verified
//
#include <hip/hip_runtime.h>

typedef float v2f __attribute__((ext_vector_type(2)));
typedef float v8f __attribute__((ext_vector_type(8)));

#define B_   1024
#define V_   6890
#define J_   24
#define NB_  10
#define C_   (V_*3)    // 20670 flat coords
#define KP_  207       // posedirs K
#define KPP_ 208       // padded K (zero col 207)
#define NT_  ((C_+15)/16)   // 1292 column tiles
#define VT_  ((V_+15)/16)   // 431 vertex tiles

// ---------------- Kernel 1: Rodrigues -> R, lrotmin (padded) ----------------
__global__ void k_rodrigues(const float* __restrict__ poses,
                            float* __restrict__ Rbuf, float* __restrict__ lrot) {
    int t = blockIdx.x * blockDim.x + threadIdx.x;
    if (t >= B_ * J_) return;
    int b = t / J_, j = t % J_;
    float rx = poses[b*(J_*3) + j*3 + 0];
    float ry = poses[b*(J_*3) + j*3 + 1];
    float rz = poses[b*(J_*3) + j*3 + 2];
    float th = sqrtf(rx*rx + ry*ry + rz*rz + 1e-8f);
    float inv = 1.0f / th;
    float ax = rx*inv, ay = ry*inv, az = rz*inv;
    float s = sinf(th), c = cosf(th), o = 1.0f - c;
    float R[9];
    R[0] = c + o*ax*ax;      R[1] = -s*az + o*ax*ay;  R[2] = s*ay + o*ax*az;
    R[3] = s*az + o*ax*ay;   R[4] = c + o*ay*ay;      R[5] = -s*ax + o*ay*az;
    R[6] = -s*ay + o*ax*az;  R[7] = s*ax + o*ay*az;   R[8] = c + o*az*az;
#pragma unroll
    for (int i = 0; i < 9; ++i) Rbuf[(b*J_ + j)*9 + i] = R[i];
    if (j > 0) {
#pragma unroll
        for (int i = 0; i < 9; ++i) {
            float id = (i == 0 || i == 4 || i == 8) ? 1.0f : 0.0f;
            lrot[b*KPP_ + (j-1)*9 + i] = R[i] - id;
        }
    } else {
        lrot[b*KPP_ + 207] = 0.0f;   // K pad column
    }
}

// ------- Kernel 2: S2T[k2][j][d] = Jreg^T @ [template | shapedirs_k] --------
__global__ void k_s2t(const float* __restrict__ vt, const float* __restrict__ sd,
                      const float* __restrict__ Jreg, float* __restrict__ S2T) {
    int o = blockIdx.x * blockDim.x + threadIdx.x;
    if (o >= 11*72) return;
    int k2 = o / 72, rem = o % 72, j = rem / 3, d = rem % 3;
    const float* src = (k2 == 0) ? vt : (sd + (k2-1)*C_);
    float acc = 0.0f;
    for (int v = 0; v < V_; ++v) acc += src[v*3 + d] * Jreg[v*J_ + j];
    S2T[o] = acc;
}

// ---------- Kernel 3: kinematic chain -> relative skinning mats A -----------
__global__ void k_kin(const float* __restrict__ shapes, const float* __restrict__ S2T,
                      const float* __restrict__ Rbuf,
                      float* __restrict__ Gbuf, float* __restrict__ Abuf) {
    int b = blockIdx.x * blockDim.x + threadIdx.x;
    if (b >= B_) return;
    float sh[NB_];
#pragma unroll
    for (int k = 0; k < NB_; ++k) sh[k] = shapes[b*NB_ + k];
    float Jl[J_*3];
#pragma unroll
    for (int e = 0; e < J_*3; ++e) {
        float acc = S2T[e];
#pragma unroll
        for (int k = 0; k < NB_; ++k) acc += sh[k] * S2T[(k+1)*72 + e];
        Jl[e] = acc;
    }
    const int PAR[J_] = {-1,0,0,0,1,2,3,4,5,6,7,8,9,9,9,12,13,14,16,17,18,19,20,21};
#pragma unroll
    for (int j = 0; j < J_; ++j) {
        float R[9];
#pragma unroll
        for (int i = 0; i < 9; ++i) R[i] = Rbuf[(b*J_ + j)*9 + i];
        float G[12];
        if (j == 0) {
#pragma unroll
            for (int m = 0; m < 3; ++m) {
                G[m*4+0] = R[m*3+0]; G[m*4+1] = R[m*3+1];
                G[m*4+2] = R[m*3+2]; G[m*4+3] = Jl[m];
            }
        } else {
            int p = PAR[j];
            float Gp[12];
#pragma unroll
            for (int i = 0; i < 12; ++i) Gp[i] = Gbuf[(b*J_ + p)*12 + i];  // same-thread RAW (same addr, ordered)
            float rel0 = Jl[j*3+0] - Jl[p*3+0];
            float rel1 = Jl[j*3+1] - Jl[p*3+1];
            float rel2 = Jl[j*3+2] - Jl[p*3+2];
#pragma unroll
            for (int m = 0; m < 3; ++m) {
#pragma unroll
                for (int n = 0; n < 3; ++n)
                    G[m*4+n] = Gp[m*4+0]*R[0*3+n] + Gp[m*4+1]*R[1*3+n] + Gp[m*4+2]*R[2*3+n];
                G[m*4+3] = Gp[m*4+0]*rel0 + Gp[m*4+1]*rel1 + Gp[m*4+2]*rel2 + Gp[m*4+3];
            }
        }
#pragma unroll
        for (int i = 0; i < 12; ++i) Gbuf[(b*J_ + j)*12 + i] = G[i];
        // A = G with translation corrected by -G_rot @ J_loc[j]; layout [24 x 16] per batch (B-matrix for WMMA)
#pragma unroll
        for (int m = 0; m < 3; ++m) {
            float tc = G[m*4+0]*Jl[j*3+0] + G[m*4+1]*Jl[j*3+1] + G[m*4+2]*Jl[j*3+2];
            Abuf[(b*J_ + j)*16 + m*4+0] = G[m*4+0];
            Abuf[(b*J_ + j)*16 + m*4+1] = G[m*4+1];
            Abuf[(b*J_ + j)*16 + m*4+2] = G[m*4+2];
            Abuf[(b*J_ + j)*16 + m*4+3] = G[m*4+3] - tc;
        }
#pragma unroll
        for (int i = 12; i < 16; ++i) Abuf[(b*J_ + j)*16 + i] = 0.0f;
    }
}

// ----- Kernel 4 (WMMA): v_posed = v_shaped + lrotmin @ posedirs -> d_out -----
__global__ void __launch_bounds__(256)
k_posegemm(const float* __restrict__ lrot, const float* __restrict__ pose,
           const float* __restrict__ shapes, const float* __restrict__ vt,
           const float* __restrict__ sd, float* __restrict__ out) {
    int tile = blockIdx.x * 8 + threadIdx.y;            // one 16x16 tile per wave
    int bt = tile / NT_, nt = tile % NT_;
    int b0 = bt * 16, n0 = nt * 16;
    int lane = threadIdx.x;
    int l16 = lane & 15, half = lane >> 4;
    int n = n0 + l16;
    int nc = (n < C_) ? n : (C_ - 1);                   // column clamp (pad cols discarded)
    const float* arow = lrot + (b0 + l16) * KPP_ + half * 2;
    v8f acc = {};
    for (int k = 0; k < KPP_; k += 4) {
        v2f a; a.x = arow[k]; a.y = arow[k + 1];        // lrotmin padded with zeros at k=207
        int kb = k + half * 2;
        int k0 = (kb     < KP_) ? kb     : (KP_ - 1);   // row clamp; A-pad=0 kills contribution
        int k1 = (kb + 1 < KP_) ? kb + 1 : (KP_ - 1);
        v2f bf; bf.x = pose[k0 * C_ + nc]; bf.y = pose[k1 * C_ + nc];
        acc = __builtin_amdgcn_wmma_f32_16x16x4_f32(false, a, false, bf,
                                                    (short)0, acc, false, false);
    }
    if (n < C_) {
        float base = vt[n];
        float sdv[NB_];
#pragma unroll
        for (int k = 0; k < NB_; ++k) sdv[k] = sd[k * C_ + n];
#pragma unroll
        for (int i = 0; i < 8; ++i) {
            int bb = b0 + half * 8 + i;                 // C/D row layout: rows 0-7 lanes<16, 8-15 lanes>=16
            float vsh = base;
#pragma unroll
            for (int k = 0; k < NB_; ++k) vsh += shapes[bb * NB_ + k] * sdv[k];
            out[bb * C_ + n] = vsh + acc[i];
        }
    }
}

// --- Kernel 5 (WMMA): T = weights @ A_b, then out = T.[v_posed,1] + trans ---
__global__ void __launch_bounds__(256)
k_skin(const float* __restrict__ w, const float* __restrict__ Abuf,
       const float* __restrict__ trans, float* __restrict__ out) {
    __shared__ float lt[8][16 * 16];
    int wave = threadIdx.y;
    int tile = blockIdx.x * 8 + wave;
    int b = tile / VT_, vtile = tile % VT_;
    int v0 = vtile * 16;
    int lane = threadIdx.x;
    int l16 = lane & 15, half = lane >> 4;
    int vA = v0 + l16; if (vA > V_ - 1) vA = V_ - 1;    // clamp loads; tail rows discarded below
    const float* arow = w + vA * J_ + half * 2;
    const float* brow = Abuf + (b * J_) * 16 + l16;
    v8f acc = {};
#pragma unroll
    for (int k = 0; k < J_; k += 4) {
        v2f a; a.x = arow[k]; a.y = arow[k + 1];
        int kb = k + half * 2;
        v2f bf; bf.x = brow[kb * 16]; bf.y = brow[(kb + 1) * 16];
        acc = __builtin_amdgcn_wmma_f32_16x16x4_f32(false, a, false, bf,
                                                    (short)0, acc, false, false);
    }
#pragma unroll
    for (int i = 0; i < 8; ++i) lt[wave][(half * 8 + i) * 16 + l16] = acc[i];
    __syncthreads();
    if (lane < 16) {
        int v = v0 + lane;
        if (v < V_) {
            float px = out[b * C_ + v * 3 + 0];         // v_posed (read before in-place write)
            float py = out[b * C_ + v * 3 + 1];
            float pz = out[b * C_ + v * 3 + 2];
#pragma unroll
            for (int m = 0; m < 3; ++m) {
                float T0 = lt[wave][lane * 16 + m * 4 + 0];
                float T1 = lt[wave][lane * 16 + m * 4 + 1];
                float T2 = lt[wave][lane * 16 + m * 4 + 2];
                float T3 = lt[wave][lane * 16 + m * 4 + 3];
                out[b * C_ + v * 3 + m] = T0 * px + T1 * py + T2 * pz + T3 + trans[b * 3 + m];
            }
        }
    }
}

extern "C" void kernel_launch(void* const* d_in, const int* in_sizes, int n_in,
                              void* d_out, int out_size, void* d_ws, size_t ws_size,
                              hipStream_t stream) {
    (void)in_sizes; (void)n_in; (void)out_size; (void)ws_size;
    const float* shapes = (const float*)d_in[0];
    const float* poses  = (const float*)d_in[1];
    const float* trans  = (const float*)d_in[2];
    const float* vt     = (const float*)d_in[3];
    const float* sd     = (const float*)d_in[4];
    const float* pd     = (const float*)d_in[5];
    const float* Jreg   = (const float*)d_in[6];
    const float* wts    = (const float*)d_in[7];
    float* out = (float*)d_out;

    float* ws   = (float*)d_ws;
    float* Rbuf = ws;                         // 1024*24*9
    float* lrot = Rbuf + B_ * J_ * 9;         // 1024*208
    float* S2T  = lrot + B_ * KPP_;           // 11*72
    float* Gbuf = S2T + 11 * 72;              // 1024*24*12
    float* Abuf = Gbuf + B_ * J_ * 12;        // 1024*24*16   (total ~4.5 MB)

    k_rodrigues<<<(B_ * J_ + 255) / 256, 256, 0, stream>>>(poses, Rbuf, lrot);
    k_s2t<<<(11 * 72 + 255) / 256, 256, 0, stream>>>(vt, sd, Jreg, S2T);
    k_kin<<<(B_ + 255) / 256, 256, 0, stream>>>(shapes, S2T, Rbuf, Gbuf, Abuf);

    dim3 blk(32, 8);
    k_posegemm<<<(64 * NT_) / 8, blk, 0, stream>>>(lrot, pd, shapes, vt, sd, out);
    k_skin<<<(B_ * VT_) / 8, blk, 0, stream>>>(wts, Abuf, trans, out);
}